// VQClassifierNNTime_26405458936338
// MI455X (gfx1250) — compile-verified
//
#include <hip/hip_runtime.h>
#include <math.h>

// ---------------- problem constants ----------------
constexpr int BT = 32768;          // 32*1024 tokens
constexpr int D  = 512;            // KEY_DIM
constexpr int D1 = 513;            // D+1
constexpr int NE = 512;            // codebook entries
constexpr int ED = 512;            // E_DIM
constexpr int KP = 520;            // padded K (513 -> 520, zero-filled tail)
constexpr int HP520 = 260;         // half-plane size for packed KP rows
constexpr int HP512 = 256;         // half-plane size for packed 512 rows
#define AK   1.3089969389957472f   // (pi/2)/RATE
#define IKT  10.0f                 // 1/KT
#define EPSN 1e-12f

typedef __attribute__((ext_vector_type(2))) float v2f;
typedef __attribute__((ext_vector_type(4))) float v4f;
typedef __attribute__((ext_vector_type(8))) float v8f;

// packed half-split K layout: lane-half h owns plane [h*HP, h*HP+HP); within a
// plane, K-quad q contributes 2 consecutive floats -> one b128 = 2 K-quads.
__device__ __forceinline__ int pidx520(int c) { return ((c >> 1) & 1) * HP520 + ((c >> 2) << 1) + (c & 1); }
__device__ __forceinline__ int pidx512(int c) { return ((c >> 1) & 1) * HP512 + ((c >> 2) << 1) + (c & 1); }

// ---------------- wave32 helpers ----------------
__device__ __forceinline__ float wsum(float v) {
#pragma unroll
  for (int m = 16; m >= 1; m >>= 1) v += __shfl_xor(v, m, 32);
  return v;
}

// 8 independent WMMAs per K-pair step (two K-quads, 32x32 output tile)
__device__ __forceinline__ void step32(const v4f a0, const v4f a1, const v4f b0, const v4f b1,
                                       v8f& c00, v8f& c01, v8f& c10, v8f& c11) {
  v2f a0l = {a0.x, a0.y}, a0h = {a0.z, a0.w};
  v2f a1l = {a1.x, a1.y}, a1h = {a1.z, a1.w};
  v2f b0l = {b0.x, b0.y}, b0h = {b0.z, b0.w};
  v2f b1l = {b1.x, b1.y}, b1h = {b1.z, b1.w};
  c00 = __builtin_amdgcn_wmma_f32_16x16x4_f32(false, a0l, false, b0l, (short)0, c00, false, false);
  c01 = __builtin_amdgcn_wmma_f32_16x16x4_f32(false, a0l, false, b1l, (short)0, c01, false, false);
  c10 = __builtin_amdgcn_wmma_f32_16x16x4_f32(false, a1l, false, b0l, (short)0, c10, false, false);
  c11 = __builtin_amdgcn_wmma_f32_16x16x4_f32(false, a1l, false, b1l, (short)0, c11, false, false);
  c00 = __builtin_amdgcn_wmma_f32_16x16x4_f32(false, a0h, false, b0h, (short)0, c00, false, false);
  c01 = __builtin_amdgcn_wmma_f32_16x16x4_f32(false, a0h, false, b1h, (short)0, c01, false, false);
  c10 = __builtin_amdgcn_wmma_f32_16x16x4_f32(false, a1h, false, b0h, (short)0, c10, false, false);
  c11 = __builtin_amdgcn_wmma_f32_16x16x4_f32(false, a1h, false, b1h, (short)0, c11, false, false);
}

// double-buffered 32x32 GEMM body over npairs b128 steps per operand row
__device__ __forceinline__ void gemm32x32(const float* __restrict__ ap0, const float* __restrict__ ap1,
                                          const float* __restrict__ bp0, const float* __restrict__ bp1,
                                          int npairs, v8f& c00, v8f& c01, v8f& c10, v8f& c11) {
  v4f a0 = *(const v4f*)ap0, a1 = *(const v4f*)ap1;
  v4f b0 = *(const v4f*)bp0, b1 = *(const v4f*)bp1;
  for (int p = 1; p < npairs; ++p) {
    v4f a0n = *(const v4f*)(ap0 + 4 * p);
    v4f a1n = *(const v4f*)(ap1 + 4 * p);
    v4f b0n = *(const v4f*)(bp0 + 4 * p);
    v4f b1n = *(const v4f*)(bp1 + 4 * p);
    step32(a0, a1, b0, b1, c00, c01, c10, c11);
    a0 = a0n; a1 = a1n; b0 = b0n; b1 = b1n;
  }
  step32(a0, a1, b0, b1, c00, c01, c10, c11);
}

// ---------------- K0: zero the idx1 histogram ----------------
__global__ void k_zero_hist(int* hist, int n) {
  int i = blockIdx.x * blockDim.x + threadIdx.x;
  if (i < n) hist[i] = 0;
}

// ---------------- K1: normalize token + time-sphere embed (packed write) ----------------
__global__ __launch_bounds__(256) void k_embed(const float* __restrict__ key_soft,
                                               const float* __restrict__ u_t,
                                               float* __restrict__ ksP,
                                               float* __restrict__ o_ks) {
  int lane = threadIdx.x & 31;
  int t = (blockIdx.x * blockDim.x + threadIdx.x) >> 5;
  const float* x = key_soft + (size_t)t * D;
  float v[16]; float ss = 0.f;
#pragma unroll
  for (int q = 0; q < 16; q++) { v[q] = x[lane + 32 * q]; ss += v[q] * v[q]; }
  ss = wsum(ss);
  float sc = 1.f / fmaxf(sqrtf(ss), EPSN);
  float a = u_t[t] * AK;
  float ca = cosf(a), sa = sinf(a);
  float* p = ksP + (size_t)t * KP;
  float* o = o_ks + (size_t)t * D1;
#pragma unroll
  for (int q = 0; q < 16; q++) {
    int c = lane + 32 * q;
    float y = v[q] * sc * ca;
    p[pidx520(c)] = y;
    o[c] = y;
  }
  if (lane < 8) p[pidx520(512 + lane)] = (lane == 0) ? sa : 0.f;
  if (lane == 0) o[512] = sa;
}

// ---------------- K2: keysA = sphere_emb(l2n(keys_w), t_keys) * clamp(r) (packed) ----------------
__global__ __launch_bounds__(256) void k_codebookA(const float* __restrict__ keys_w,
                                                   const float* __restrict__ t_keys,
                                                   const float* __restrict__ r_keys,
                                                   float* __restrict__ kA) {
  int lane = threadIdx.x & 31;
  int i = (blockIdx.x * blockDim.x + threadIdx.x) >> 5;
  const float* x = keys_w + (size_t)i * D;
  float v[16]; float ss = 0.f;
#pragma unroll
  for (int q = 0; q < 16; q++) { v[q] = x[lane + 32 * q]; ss += v[q] * v[q]; }
  ss = wsum(ss);
  float sc = 1.f / fmaxf(sqrtf(ss), EPSN);
  float a = t_keys[i] * AK;
  float ca = cosf(a), sa = sinf(a);
  float r = fminf(fmaxf(r_keys[i], 0.f), 1.f);
  float* p = kA + (size_t)i * KP;
#pragma unroll
  for (int q = 0; q < 16; q++) { int c = lane + 32 * q; p[pidx520(c)] = v[q] * sc * ca * r; }
  if (lane < 8) p[pidx520(512 + lane)] = (lane == 0) ? sa * r : 0.f;
}

// ---------------- K3: score0 with fused argmax; 32 tokens x all codes per wave ----------------
__global__ __launch_bounds__(128) void k_score0_argmax(const float* __restrict__ ksP,
                                                       const float* __restrict__ kA,
                                                       int* __restrict__ idx0) {
  int lane = threadIdx.x & 31;
  int l15 = lane & 15, half = lane >> 4;
  int wave = (blockIdx.x * blockDim.x + threadIdx.x) >> 5;  // 0..1023
  int m0 = wave * 32;
  const float* ap0 = ksP + (size_t)(m0 + l15) * KP + half * HP520;
  const float* ap1 = ksP + (size_t)(m0 + 16 + l15) * KP + half * HP520;
  float blo[8], bhi[8]; int ilo[8], ihi[8];
#pragma unroll
  for (int r = 0; r < 8; r++) { blo[r] = bhi[r] = -1e30f; ilo[r] = ihi[r] = 0; }
  for (int n0 = 0; n0 < NE; n0 += 32) {
    const float* bp0 = kA + (size_t)(n0 + l15) * KP + half * HP520;
    const float* bp1 = kA + (size_t)(n0 + 16 + l15) * KP + half * HP520;
    v8f c00 = {}, c01 = {}, c10 = {}, c11 = {};
    gemm32x32(ap0, ap1, bp0, bp1, HP520 / 4, c00, c01, c10, c11);
#pragma unroll
    for (int r = 0; r < 8; r++) {
      // low 16 tokens: candidates from c00 (cols n0+l15) and c01 (cols n0+16+l15)
      float v = c00[r]; int vi = n0 + l15;
      float v2 = c01[r]; int vi2 = n0 + 16 + l15;
      if (v2 > v) { v = v2; vi = vi2; }  // tie keeps lower (c00) index
#pragma unroll
      for (int m = 8; m >= 1; m >>= 1) {
        float ov = __shfl_xor(v, m, 32); int oi = __shfl_xor(vi, m, 32);
        if (ov > v || (ov == v && oi < vi)) { v = ov; vi = oi; }
      }
      if (v > blo[r] || (v == blo[r] && vi < ilo[r])) { blo[r] = v; ilo[r] = vi; }
      // high 16 tokens: c10 / c11
      float u = c10[r]; int ui = n0 + l15;
      float u2 = c11[r]; int ui2 = n0 + 16 + l15;
      if (u2 > u) { u = u2; ui = ui2; }
#pragma unroll
      for (int m = 8; m >= 1; m >>= 1) {
        float ou = __shfl_xor(u, m, 32); int oi = __shfl_xor(ui, m, 32);
        if (ou > u || (ou == u && oi < ui)) { u = ou; ui = oi; }
      }
      if (u > bhi[r] || (u == bhi[r] && ui < ihi[r])) { bhi[r] = u; ihi[r] = ui; }
    }
  }
  if (l15 == 0) {
#pragma unroll
    for (int r = 0; r < 8; r++) {
      idx0[m0 + r + half * 8] = ilo[r];
      idx0[m0 + 16 + r + half * 8] = ihi[r];
    }
  }
}

// ---------------- K4: deterministic segment sums over packed rows ----------------
__global__ __launch_bounds__(256) void k_seg_sum(const float* __restrict__ ksP,
                                                 const int* __restrict__ idx0,
                                                 float* __restrict__ seg,
                                                 float* __restrict__ cnt) {
  int code = blockIdx.x;
  int tid = threadIdx.x;
  float a0 = 0.f, a1 = 0.f, a2 = 0.f, c = 0.f;
  for (int t = 0; t < BT; t++) {
    if (idx0[t] == code) {
      const float* row = ksP + (size_t)t * KP;
      a0 += row[tid];
      a1 += row[tid + 256];
      if (tid < 8) a2 += row[512 + tid];
      if (tid == 0) c += 1.f;
    }
  }
  float* s = seg + (size_t)code * KP;
  s[tid] = a0; s[tid + 256] = a1;
  if (tid < 8) s[512 + tid] = a2;
  if (tid == 0) cnt[code] = c;
}

// ---------------- K5: EMA update -> (t_new, feat_new); packed slot 256 = sine ----------------
__global__ __launch_bounds__(256) void k_keys_update(const float* __restrict__ kA,
                                                     const float* __restrict__ seg,
                                                     const float* __restrict__ cnt,
                                                     float* __restrict__ feat,
                                                     float* __restrict__ tnew) {
  int lane = threadIdx.x & 31;
  int i = (blockIdx.x * blockDim.x + threadIdx.x) >> 5;
  const float* s = seg + (size_t)i * KP;
  const float* ka = kA + (size_t)i * KP;
  float cinv = 1.0f / (1.0f + cnt[i]);
  float tmp[16]; float ss = 0.f;
#pragma unroll
  for (int q = 0; q < 16; q++) {
    int d = lane + 32 * q;
    float v = ka[d] * 0.5f + (s[d] * cinv) * 0.5f;
    tmp[q] = v;
    ss += (d == 256) ? 0.f : v * v;  // exclude the sine slot from feature norm
  }
  float ve = 0.f;
  if (lane < 8) { int d = 512 + lane; ve = ka[d] * 0.5f + (s[d] * cinv) * 0.5f; }  // lanes 0-3 data, 4-7 zero pads
  float red512 = wsum(ss);
  float lb = __shfl(tmp[8], 0, 32);  // packed slot 256 lives at lane 0, q=8
  float red513 = red512 + lb * lb;
  float n513 = fmaxf(sqrtf(red513), EPSN);
  float sN = fminf(fmaxf(lb / n513, -1.f), 1.f);
  if (lane == 0) tnew[i] = asinf(sN) / AK;
  float s512 = 1.0f / fmaxf(sqrtf(red512), EPSN);
  float* fo = feat + (size_t)i * KP;
#pragma unroll
  for (int q = 0; q < 16; q++) {
    int d = lane + 32 * q;
    fo[d] = (d == 256) ? 0.f : tmp[q] * s512;
  }
  if (lane < 8) fo[512 + lane] = ve * s512;  // lanes 0-3 data, 4-7 zeros
}

// ---------------- K6: bitonic argsort of t_new (stable via index tiebreak) ----------------
__global__ void k_sort(const float* __restrict__ tnew,
                       float* __restrict__ tsort, int* __restrict__ isw) {
  __shared__ float sv[NE];
  __shared__ int si[NE];
  int tid = threadIdx.x;
  sv[tid] = tnew[tid]; si[tid] = tid;
  __syncthreads();
  for (int k = 2; k <= NE; k <<= 1)
    for (int j = k >> 1; j > 0; j >>= 1) {
      int ixj = tid ^ j;
      if (ixj > tid) {
        bool up = ((tid & k) == 0);
        float a = sv[tid], b = sv[ixj];
        int ia = si[tid], ib = si[ixj];
        bool agtb = (a > b) || (a == b && ia > ib);
        if (up ? agtb : !agtb) { sv[tid] = b; sv[ixj] = a; si[tid] = ib; si[ixj] = ia; }
      }
      __syncthreads();
    }
  tsort[tid] = sv[tid]; isw[tid] = si[tid];
}

// ---------------- K7: build packed keysB and packed-K vpnT ----------------
__global__ __launch_bounds__(256) void k_buildB(const float* __restrict__ feat,
                                                const float* __restrict__ tsort,
                                                const int* __restrict__ isw,
                                                const float* __restrict__ r_keys,
                                                const float* __restrict__ vparams,
                                                float* __restrict__ kB,
                                                float* __restrict__ vpnT) {
  int lane = threadIdx.x & 31;
  int i = (blockIdx.x * blockDim.x + threadIdx.x) >> 5;  // sorted code index
  int sw = isw[i];
  const float* f = feat + (size_t)sw * KP;  // packed row; slot 256 and pads are 0
  float v[16]; float ss = 0.f;
#pragma unroll
  for (int q = 0; q < 16; q++) { v[q] = f[lane + 32 * q]; ss += v[q] * v[q]; }
  float ve = (lane < 8) ? f[512 + lane] : 0.f;  // lanes 0-3 data, 4-7 zeros
  ss = wsum(ss + ve * ve);
  float sc = 1.f / fmaxf(sqrtf(ss), EPSN);
  float a = tsort[i] * AK;
  float ca = cosf(a), sa = sinf(a);
  float r2 = fminf(fmaxf(r_keys[sw], 0.f), 1.f);
  float* p = kB + (size_t)i * KP;
#pragma unroll
  for (int q = 0; q < 16; q++) p[lane + 32 * q] = v[q] * sc * ca * r2;
  if (lane < 8) p[512 + lane] = ve * sc * ca * r2;  // lanes 0-3 data, 4-7 zeros
  if (lane == 0) p[256] = sa * r2;                  // sine slot (overwrites the zero)
  // vpnT[e][pidx512(i)] = split_norm(vparams[sw])[e]   (K = code axis packed)
  const float* vp = vparams + (size_t)sw * ED;
  int pi = pidx512(i);
#pragma unroll
  for (int c = 0; c < 8; c++) {
    float y0 = vp[c * 64 + lane], y1 = vp[c * 64 + lane + 32];
    float s2 = wsum(y0 * y0 + y1 * y1);
    float s = 1.f / fmaxf(sqrtf(s2), EPSN);
    vpnT[(size_t)(c * 64 + lane) * NE + pi] = y0 * s;
    vpnT[(size_t)(c * 64 + lane + 32) * NE + pi] = y1 * s;
  }
}

// ---------------- K8: idx1 = isw[idx0]; histogram of idx1 ----------------
__global__ void k_idx1(const int* __restrict__ idx0, const int* __restrict__ isw,
                       int* __restrict__ idx1, int* __restrict__ hist) {
  int t = blockIdx.x * blockDim.x + threadIdx.x;
  if (t < BT) {
    int j = isw[idx0[t]];
    idx1[t] = j;
    __hip_atomic_fetch_add(hist + j, 1, __ATOMIC_RELAXED, __HIP_MEMORY_SCOPE_AGENT);
  }
}

// ---------------- K9: score = ksP @ kB^T, 32x32 tile per wave, plain-N stores ----------------
__global__ __launch_bounds__(128) void k_score2(const float* __restrict__ ksP,
                                                const float* __restrict__ kB,
                                                float* __restrict__ score) {
  int lane = threadIdx.x & 31;
  int l15 = lane & 15, half = lane >> 4;
  int wave = (blockIdx.x * blockDim.x + threadIdx.x) >> 5;  // 0..16383
  int m0 = (wave >> 4) * 32;
  int n0 = (wave & 15) * 32;
  const float* ap0 = ksP + (size_t)(m0 + l15) * KP + half * HP520;
  const float* ap1 = ksP + (size_t)(m0 + 16 + l15) * KP + half * HP520;
  const float* bp0 = kB + (size_t)(n0 + l15) * KP + half * HP520;
  const float* bp1 = kB + (size_t)(n0 + 16 + l15) * KP + half * HP520;
  v8f c00 = {}, c01 = {}, c10 = {}, c11 = {};
  gemm32x32(ap0, ap1, bp0, bp1, HP520 / 4, c00, c01, c10, c11);
#pragma unroll
  for (int r = 0; r < 8; r++) {
    size_t rlo = (size_t)(m0 + r + half * 8) * NE;
    size_t rhi = (size_t)(m0 + 16 + r + half * 8) * NE;
    score[rlo + n0 + l15] = c00[r];
    score[rlo + n0 + 16 + l15] = c01[r];
    score[rhi + n0 + l15] = c10[r];
    score[rhi + n0 + 16 + l15] = c11[r];
  }
}

// ---------------- K10: in-place row softmax; writes w re-packed along N ----------------
__global__ __launch_bounds__(256) void k_softmax(float* __restrict__ score) {
  int lane = threadIdx.x & 31;
  int t = (blockIdx.x * blockDim.x + threadIdx.x) >> 5;
  float* row = score + (size_t)t * NE;
  float v[16]; float mx = -1e30f;
#pragma unroll
  for (int q = 0; q < 16; q++) { v[q] = row[lane + 32 * q]; mx = fmaxf(mx, v[q]); }
#pragma unroll
  for (int m = 16; m >= 1; m >>= 1) mx = fmaxf(mx, __shfl_xor(mx, m, 32));
  float sum = 0.f;
#pragma unroll
  for (int q = 0; q < 16; q++) { v[q] = expf((v[q] - mx) * IKT); sum += v[q]; }
  sum = wsum(sum);
  float inv = 1.0f / sum;  // sum >= 1 (max element), never NaN
  // all reads above complete before any store below (single wave owns the row)
#pragma unroll
  for (int q = 0; q < 16; q++) row[pidx512(lane + 32 * q)] = v[q] * inv;
}

// ---------------- K11: vp_w_raw = w @ vpn (both K-packed), 32x32 tiles ----------------
__global__ __launch_bounds__(128) void k_vpw_gemm(const float* __restrict__ w,
                                                  const float* __restrict__ vpnT,
                                                  float* __restrict__ vpw) {
  int lane = threadIdx.x & 31;
  int l15 = lane & 15, half = lane >> 4;
  int wave = (blockIdx.x * blockDim.x + threadIdx.x) >> 5;
  int m0 = (wave >> 4) * 32;
  int n0 = (wave & 15) * 32;
  const float* ap0 = w + (size_t)(m0 + l15) * NE + half * HP512;
  const float* ap1 = w + (size_t)(m0 + 16 + l15) * NE + half * HP512;
  const float* bp0 = vpnT + (size_t)(n0 + l15) * NE + half * HP512;
  const float* bp1 = vpnT + (size_t)(n0 + 16 + l15) * NE + half * HP512;
  v8f c00 = {}, c01 = {}, c10 = {}, c11 = {};
  gemm32x32(ap0, ap1, bp0, bp1, HP512 / 4, c00, c01, c10, c11);
#pragma unroll
  for (int r = 0; r < 8; r++) {
    size_t rlo = (size_t)(m0 + r + half * 8) * ED;
    size_t rhi = (size_t)(m0 + 16 + r + half * 8) * ED;
    vpw[rlo + n0 + l15] = c00[r];
    vpw[rlo + n0 + 16 + l15] = c01[r];
    vpw[rhi + n0 + l15] = c10[r];
    vpw[rhi + n0 + 16 + l15] = c11[r];
  }
}

// ---------------- K12: epilogue — gathers, split_norms, scalar outputs ----------------
__global__ __launch_bounds__(256) void k_epilogue(const float* __restrict__ kB,
                                                  const float* __restrict__ w,
                                                  const float* __restrict__ vpw,
                                                  const int* __restrict__ idx1,
                                                  const int* __restrict__ isw,
                                                  const int* __restrict__ hist,
                                                  const float* __restrict__ vparams,
                                                  float* __restrict__ o_idx,
                                                  float* __restrict__ o_keyh,
                                                  float* __restrict__ o_vpw,
                                                  float* __restrict__ o_vph,
                                                  float* __restrict__ o_wmax,
                                                  float* __restrict__ o_wcnt) {
  int lane = threadIdx.x & 31;
  int t = (blockIdx.x * blockDim.x + threadIdx.x) >> 5;
  int j = idx1[t];
  if (lane == 0) {
    o_idx[t] = (float)j;
    o_wmax[t] = w[(size_t)t * NE + pidx512(j)];  // w rows are N-packed
    o_wcnt[t] = 1.0f / (float)(hist[j] + 1);
  }
  // key_hard = keysB[j] (straight-through cancels to the hard value); un-permute
  const float* kb = kB + (size_t)j * KP;
  float* kh = o_keyh + (size_t)t * D1;
#pragma unroll
  for (int q = 0; q < 16; q++) { int c = lane + 32 * q; kh[c] = kb[pidx520(c)]; }
  if (lane == 0) kh[512] = kb[256];
  // vp_w = split_norm(vpw row)  (plain layout)
  const float* vr = vpw + (size_t)t * ED;
  float* ov = o_vpw + (size_t)t * ED;
#pragma unroll
  for (int c = 0; c < 8; c++) {
    float y0 = vr[c * 64 + lane], y1 = vr[c * 64 + lane + 32];
    float s2 = wsum(y0 * y0 + y1 * y1);
    float s = 1.f / fmaxf(sqrtf(s2), EPSN);
    ov[c * 64 + lane] = y0 * s;
    ov[c * 64 + lane + 32] = y1 * s;
  }
  // vparams_hard = split_norm(vparams[isw[j]])
  const float* vp = vparams + (size_t)isw[j] * ED;
  float* oh = o_vph + (size_t)t * ED;
#pragma unroll
  for (int c = 0; c < 8; c++) {
    float y0 = vp[c * 64 + lane], y1 = vp[c * 64 + lane + 32];
    float s2 = wsum(y0 * y0 + y1 * y1);
    float s = 1.f / fmaxf(sqrtf(s2), EPSN);
    oh[c * 64 + lane] = y0 * s;
    oh[c * 64 + lane + 32] = y1 * s;
  }
}

// ---------------- host-side launch ----------------
extern "C" void kernel_launch(void* const* d_in, const int* in_sizes, int n_in,
                              void* d_out, int out_size, void* d_ws, size_t ws_size,
                              hipStream_t stream) {
  (void)in_sizes; (void)n_in; (void)out_size; (void)ws_size;
  const float* key_soft = (const float*)d_in[0];
  const float* u_t      = (const float*)d_in[1];
  const float* keys_w   = (const float*)d_in[2];
  const float* t_keys   = (const float*)d_in[3];
  const float* r_keys   = (const float*)d_in[4];
  const float* vparams  = (const float*)d_in[5];
  float* out = (float*)d_out;

  // workspace layout (floats; ~208 MB total)
  float* ws    = (float*)d_ws;
  float* ksP   = ws;                              // BT*KP   (packed)
  float* kA    = ksP + (size_t)BT * KP;           // NE*KP   (packed)
  float* kB    = kA + (size_t)NE * KP;            // NE*KP   (packed)
  float* seg   = kB + (size_t)NE * KP;            // NE*KP   (packed)
  float* cnt   = seg + (size_t)NE * KP;           // NE
  float* feat  = cnt + NE;                        // NE*KP   (packed)
  float* tnew  = feat + (size_t)NE * KP;          // NE
  float* tsort = tnew + NE;                       // NE
  float* vpnT  = tsort + NE;                      // ED*NE   (K-packed)
  float* score = vpnT + (size_t)ED * NE;          // BT*NE   (w after softmax, N-packed)
  float* vpw   = score + (size_t)BT * NE;         // BT*ED
  int* ints = (int*)(vpw + (size_t)BT * ED);
  int* idx0 = ints;                               // BT
  int* idx1 = idx0 + BT;                          // BT
  int* isw  = idx1 + BT;                          // NE
  int* hist = isw + NE;                           // NE

  // output layout (tuple, flat, return order)
  float* o_idx  = out;                            // BT
  float* o_keyh = o_idx + BT;                     // BT*D1
  float* o_vpw  = o_keyh + (size_t)BT * D1;       // BT*ED
  float* o_vph  = o_vpw + (size_t)BT * ED;        // BT*ED
  float* o_wmax = o_vph + (size_t)BT * ED;        // BT
  float* o_wcnt = o_wmax + BT;                    // BT
  float* o_ks   = o_wcnt + BT;                    // BT*D1

  k_zero_hist<<<2, 256, 0, stream>>>(hist, NE);
  k_embed<<<4096, 256, 0, stream>>>(key_soft, u_t, ksP, o_ks);
  k_codebookA<<<64, 256, 0, stream>>>(keys_w, t_keys, r_keys, kA);
  k_score0_argmax<<<256, 128, 0, stream>>>(ksP, kA, idx0);
  k_seg_sum<<<NE, 256, 0, stream>>>(ksP, idx0, seg, cnt);
  k_keys_update<<<64, 256, 0, stream>>>(kA, seg, cnt, feat, tnew);
  k_sort<<<1, NE, 0, stream>>>(tnew, tsort, isw);
  k_buildB<<<64, 256, 0, stream>>>(feat, tsort, isw, r_keys, vparams, kB, vpnT);
  k_idx1<<<128, 256, 0, stream>>>(idx0, isw, idx1, hist);
  k_score2<<<4096, 128, 0, stream>>>(ksP, kB, score);
  k_softmax<<<4096, 256, 0, stream>>>(score);
  k_vpw_gemm<<<4096, 128, 0, stream>>>(score, vpnT, vpw);
  k_epilogue<<<4096, 256, 0, stream>>>(kB, score, vpw, idx1, isw, hist, vparams,
                                       o_idx, o_keyh, o_vpw, o_vph, o_wmax, o_wcnt);
}